// GraphAttentionLayer_85572928406097
// MI455X (gfx1250) — compile-verified
//
#include <hip/hip_runtime.h>
#include <cmath>

#define N_NODES 50000
#define E_EDGES 800000
#define IN_F    128
#define OUT_F   64
#define LRELU_ALPHA 0.2f

typedef float v2f __attribute__((ext_vector_type(2)));
typedef float v8f __attribute__((ext_vector_type(8)));

// ---------------------------------------------------------------------------
// Kernel 1: Wh = h @ W via V_WMMA_F32_16X16X4_F32 (f32 WMMA, full precision).
// One wave computes a 16x64 output tile (4 accumulators of 16x16).
// N = 50000 = 3125 * 16 exactly -> no ragged rows, EXEC stays all-ones.
// A-frag layout (16x4 f32, ISA 7.12.2): lane L holds m=L%16, K = 2*(L/16)+v.
// B-frag layout (4x16):                 lane L holds n=L%16, K = 2*(L/16)+v.
// C/D layout: VGPR r -> row r (lanes 0-15) / row r+8 (lanes 16-31), col = L%16.
// ---------------------------------------------------------------------------
__global__ __launch_bounds__(256) void gat_gemm_wh(const float* __restrict__ h,
                                                   const float* __restrict__ W,
                                                   float* __restrict__ Wh) {
    const int lane     = threadIdx.x & 31;
    const int wave     = threadIdx.x >> 5;
    const int row_tile = blockIdx.x * 8 + wave;          // 16 rows per tile
    if (row_tile * 16 >= N_NODES) return;                // uniform per wave

    const int n   = lane & 15;                            // col-in-tile / row-in-tile
    const int kh  = (lane >> 4) << 1;                     // 0 or 2
    const int row = row_tile * 16 + n;                    // A row for this lane

    v8f acc0 = {}, acc1 = {}, acc2 = {}, acc3 = {};
    const float* __restrict__ hrow = h + (size_t)row * IN_F;

#pragma unroll 4
    for (int kb = 0; kb < IN_F; kb += 4) {
        v2f a;
        a.x = hrow[kb + kh];
        a.y = hrow[kb + kh + 1];

        const float* __restrict__ wk = W + (size_t)(kb + kh) * OUT_F + n;
        v2f b0, b1, b2, b3;
        b0.x = wk[0];   b0.y = wk[OUT_F + 0];
        b1.x = wk[16];  b1.y = wk[OUT_F + 16];
        b2.x = wk[32];  b2.y = wk[OUT_F + 32];
        b3.x = wk[48];  b3.y = wk[OUT_F + 48];

        acc0 = __builtin_amdgcn_wmma_f32_16x16x4_f32(false, a, false, b0, (short)0, acc0, false, false);
        acc1 = __builtin_amdgcn_wmma_f32_16x16x4_f32(false, a, false, b1, (short)0, acc1, false, false);
        acc2 = __builtin_amdgcn_wmma_f32_16x16x4_f32(false, a, false, b2, (short)0, acc2, false, false);
        acc3 = __builtin_amdgcn_wmma_f32_16x16x4_f32(false, a, false, b3, (short)0, acc3, false, false);
    }

    const int mbase = row_tile * 16 + ((lane >> 4) << 3); // +8 rows for upper lanes
#pragma unroll
    for (int r = 0; r < 8; ++r) {
        float* __restrict__ orow = Wh + (size_t)(mbase + r) * OUT_F + n;
        orow[0]  = acc0[r];
        orow[16] = acc1[r];
        orow[32] = acc2[r];
        orow[48] = acc3[r];
    }
}

// ---------------------------------------------------------------------------
// Kernel 2: s1[i] = Wh[i,:] . a[0:64],  s2[i] = Wh[i,:] . a[64:128]
// ---------------------------------------------------------------------------
__global__ void gat_node_scores(const float* __restrict__ Wh,
                                const float* __restrict__ a,
                                float* __restrict__ s1, float* __restrict__ s2) {
    const int i = blockIdx.x * blockDim.x + threadIdx.x;
    if (i >= N_NODES) return;
    const float* __restrict__ w = Wh + (size_t)i * OUT_F;
    float acc1 = 0.f, acc2 = 0.f;
#pragma unroll 8
    for (int f = 0; f < OUT_F; ++f) {
        const float v = w[f];
        acc1 = fmaf(v, a[f], acc1);
        acc2 = fmaf(v, a[OUT_F + f], acc2);
    }
    s1[i] = acc1;
    s2[i] = acc2;
}

// ---------------------------------------------------------------------------
// Kernel 3: zero accumulation buffers (d_out and denom) -- ws/d_out are
// poisoned by the harness and not re-zeroed between replays.
// ---------------------------------------------------------------------------
__global__ void gat_zero(float* __restrict__ out, float* __restrict__ denom) {
    const int i = blockIdx.x * blockDim.x + threadIdx.x;
    if (i < N_NODES * OUT_F) out[i] = 0.f;
    if (i < N_NODES)         denom[i] = 0.f;
}

// ---------------------------------------------------------------------------
// Kernel 4: per-edge: e = leaky_relu(s1[src]+s2[dst]); exp; segment-sum denom.
// edge_index dtype is int64 per the reference.
// ---------------------------------------------------------------------------
__global__ void gat_edge_scores(const long long* __restrict__ src,
                                const long long* __restrict__ dst,
                                const float* __restrict__ s1,
                                const float* __restrict__ s2,
                                float* __restrict__ exp_e,
                                float* __restrict__ denom) {
    const int e = blockIdx.x * blockDim.x + threadIdx.x;
    if (e >= E_EDGES) return;
    const int s = (int)src[e];
    const int d = (int)dst[e];
    float x = s1[s] + s2[d];
    x = (x > 0.f) ? x : LRELU_ALPHA * x;
    const float ex = expf(x);
    exp_e[e] = ex;
    atomicAdd(&denom[d], ex);
}

// ---------------------------------------------------------------------------
// Kernel 5: out[dst] += (exp_e/denom[dst]) * Wh[src].  One wave per edge,
// 2 feats per lane (float2 gather from L2-resident Wh, f32 atomics into L2).
// ---------------------------------------------------------------------------
__global__ __launch_bounds__(256) void gat_edge_scatter(const long long* __restrict__ src,
                                                        const long long* __restrict__ dst,
                                                        const float* __restrict__ exp_e,
                                                        const float* __restrict__ denom,
                                                        const float* __restrict__ Wh,
                                                        float* __restrict__ out) {
    const int lane = threadIdx.x & 31;
    const int e    = blockIdx.x * 8 + (threadIdx.x >> 5);
    if (e >= E_EDGES) return;
    const int s = (int)src[e];
    const int d = (int)dst[e];
    const float att = exp_e[e] / denom[d];
    const int f = lane << 1;
    const float2 wh = *(const float2*)(Wh + (size_t)s * OUT_F + f);
    float* __restrict__ o = out + (size_t)d * OUT_F + f;
    atomicAdd(o,     att * wh.x);
    atomicAdd(o + 1, att * wh.y);
}

// ---------------------------------------------------------------------------
// Kernel 6: ELU in place on d_out.
// ---------------------------------------------------------------------------
__global__ void gat_elu(float* __restrict__ out) {
    const int i = blockIdx.x * blockDim.x + threadIdx.x;
    if (i >= N_NODES * OUT_F) return;
    const float v = out[i];
    out[i] = (v > 0.f) ? v : expm1f(v);
}

// ---------------------------------------------------------------------------
// Launch
// ---------------------------------------------------------------------------
extern "C" void kernel_launch(void* const* d_in, const int* in_sizes, int n_in,
                              void* d_out, int out_size, void* d_ws, size_t ws_size,
                              hipStream_t stream) {
    const float*     h    = (const float*)d_in[0];
    const long long* eidx = (const long long*)d_in[1];  // [2, E] int64
    const float*     W    = (const float*)d_in[2];
    const float*     a    = (const float*)d_in[3];
    float*           out  = (float*)d_out;

    const long long* src = eidx;
    const long long* dst = eidx + E_EDGES;

    // Workspace carve-out (256B aligned slabs)
    char* ws = (char*)d_ws;
    size_t off = 0;
    auto alloc = [&](size_t bytes) {
        char* p = ws + off;
        off += (bytes + 255) & ~(size_t)255;
        return p;
    };
    float* Wh    = (float*)alloc((size_t)N_NODES * OUT_F * sizeof(float)); // 12.8 MB
    float* s1    = (float*)alloc((size_t)N_NODES * sizeof(float));
    float* s2    = (float*)alloc((size_t)N_NODES * sizeof(float));
    float* exp_e = (float*)alloc((size_t)E_EDGES * sizeof(float));         // 3.2 MB
    float* denom = (float*)alloc((size_t)N_NODES * sizeof(float));
    (void)ws_size; (void)in_sizes; (void)n_in; (void)out_size;

    // 1) WMMA GEMM: Wh = h @ W   (3125 row-tiles, 8 waves/block)
    const int row_tiles = N_NODES / 16;                   // 3125
    gat_gemm_wh<<<(row_tiles + 7) / 8, 256, 0, stream>>>(h, W, Wh);

    // 2) node attention scores
    gat_node_scores<<<(N_NODES + 255) / 256, 256, 0, stream>>>(Wh, a, s1, s2);

    // 3) zero accumulators
    gat_zero<<<(N_NODES * OUT_F + 255) / 256, 256, 0, stream>>>(out, denom);

    // 4) edge scores + softmax denominators
    gat_edge_scores<<<(E_EDGES + 255) / 256, 256, 0, stream>>>(src, dst, s1, s2, exp_e, denom);

    // 5) weighted scatter (1 wave per edge)
    gat_edge_scatter<<<E_EDGES / 8, 256, 0, stream>>>(src, dst, exp_e, denom, Wh, out);

    // 6) ELU epilogue
    gat_elu<<<(N_NODES * OUT_F + 255) / 256, 256, 0, stream>>>(out);
}